// FarthestPointSampling_87050397155539
// MI455X (gfx1250) — compile-verified
//
#include <hip/hip_runtime.h>

// Farthest Point Sampling for MI455X (gfx1250, wave32).
// One 1024-thread workgroup per batch (B=8). min_d lives on-chip:
//   - 64/thread in a fully-unrolled float4[16] register array
//   - 65536 entries in LDS (256 KB) accessed as b128
// Points stream from L2 as global_load_b128; winner-coordinate broadcast
// uses global_load_async_to_lds_b32 (ASYNCcnt) on the critical path.

#define TPB    1024
#define NPTS   131072
#define LDSN   65536            // min_d entries in LDS (256 KB)
#define REGG   16               // float4 groups/thread, register half (64 pts)
#define LDSG   16               // float4 groups/thread, LDS half (64 pts)
#define NWAVES (TPB/32)

__device__ __forceinline__ void take_better(float& bv, int& bi, float ov, int oi) {
    if (ov > bv || (ov == bv && oi < bi)) { bv = ov; bi = oi; }
}

__global__ __launch_bounds__(TPB, 1)
void fps_kernel(const float* __restrict__ points, float* __restrict__ out,
                int N, int S) {
    __shared__ float4 s_mind4[LDSN / 4];
    __shared__ float  s_redv[NWAVES];
    __shared__ int    s_redi[NWAVES];
    __shared__ float  s_cur[4];

    const int b = blockIdx.x;
    const int t = threadIdx.x;
    const float* __restrict__ px = points + (size_t)b * 3 * N;
    const float* __restrict__ py = px + N;
    const float* __restrict__ pz = px + 2 * N;
    const float4* __restrict__ px4 = (const float4*)px;
    const float4* __restrict__ py4 = (const float4*)py;
    const float4* __restrict__ pz4 = (const float4*)pz;
    float* __restrict__ outb = out + (size_t)b * 3 * S;

    // Warm L2 with this batch's points (global_prefetch_b8, 128B granules).
    for (int off = t * 32; off < 3 * NPTS; off += TPB * 32)
        __builtin_prefetch(px + off, 0, 1);

    // Init running min distances.
    float4 md[REGG];
#pragma unroll
    for (int k = 0; k < REGG; ++k)
        md[k] = make_float4(1e10f, 1e10f, 1e10f, 1e10f);
#pragma unroll 4
    for (int k = 0; k < LDSG; ++k)
        s_mind4[k * TPB + t] = make_float4(1e10f, 1e10f, 1e10f, 1e10f);
    if (t == 0) {                       // first selected index is 0
        s_cur[0] = px[0]; s_cur[1] = py[0]; s_cur[2] = pz[0];
    }

    for (int s = 0; s < S; ++s) {
        __syncthreads();                // s_cur / s_mind visible
        const float cx = s_cur[0], cy = s_cur[1], cz = s_cur[2];
        if (t == 0) {                   // emit coords of current index
            outb[s] = cx; outb[S + s] = cy; outb[2 * S + s] = cz;
        }

        float bv = -1.0f;               // local best (value, index)
        int   bi = 0x7fffffff;

        // --- register-resident half: groups [0, REGG*TPB), points [0, 65536) ---
#pragma unroll
        for (int k = 0; k < REGG; ++k) {
            const int g = k * TPB + t;
            const int j = g * 4;
            const float4 x = px4[g], y = py4[g], z = pz4[g];
            float4 m = md[k];
            float dx, dy, dz, d;
            dx = x.x - cx; dy = y.x - cy; dz = z.x - cz; d = dx*dx + dy*dy + dz*dz;
            m.x = fminf(m.x, d); take_better(bv, bi, m.x, j + 0);
            dx = x.y - cx; dy = y.y - cy; dz = z.y - cz; d = dx*dx + dy*dy + dz*dz;
            m.y = fminf(m.y, d); take_better(bv, bi, m.y, j + 1);
            dx = x.z - cx; dy = y.z - cy; dz = z.z - cz; d = dx*dx + dy*dy + dz*dz;
            m.z = fminf(m.z, d); take_better(bv, bi, m.z, j + 2);
            dx = x.w - cx; dy = y.w - cy; dz = z.w - cz; d = dx*dx + dy*dy + dz*dz;
            m.w = fminf(m.w, d); take_better(bv, bi, m.w, j + 3);
            md[k] = m;
        }
        // --- LDS-resident half: groups [REGG*TPB, 2*REGG*TPB), points [65536, 131072) ---
#pragma unroll 4
        for (int k = 0; k < LDSG; ++k) {
            const int la = k * TPB + t;
            const int g  = (REGG + k) * TPB + t;
            const int j  = g * 4;
            const float4 x = px4[g], y = py4[g], z = pz4[g];
            float4 m = s_mind4[la];     // ds_load_b128
            float dx, dy, dz, d;
            dx = x.x - cx; dy = y.x - cy; dz = z.x - cz; d = dx*dx + dy*dy + dz*dz;
            m.x = fminf(m.x, d); take_better(bv, bi, m.x, j + 0);
            dx = x.y - cx; dy = y.y - cy; dz = z.y - cz; d = dx*dx + dy*dy + dz*dz;
            m.y = fminf(m.y, d); take_better(bv, bi, m.y, j + 1);
            dx = x.z - cx; dy = y.z - cy; dz = z.z - cz; d = dx*dx + dy*dy + dz*dz;
            m.z = fminf(m.z, d); take_better(bv, bi, m.z, j + 2);
            dx = x.w - cx; dy = y.w - cy; dz = z.w - cz; d = dx*dx + dy*dy + dz*dz;
            m.w = fminf(m.w, d); take_better(bv, bi, m.w, j + 3);
            s_mind4[la] = m;            // ds_store_b128
        }

        // --- wave32 argmax butterfly (first-index tie-break) ---
#pragma unroll
        for (int off = 16; off > 0; off >>= 1) {
            const float ov = __shfl_xor(bv, off, 32);
            const int   oi = __shfl_xor(bi, off, 32);
            take_better(bv, bi, ov, oi);
        }
        const int wv = t >> 5;
        if ((t & 31) == 0) { s_redv[wv] = bv; s_redi[wv] = bi; }
        __syncthreads();

        // --- cross-wave reduce in wave 0 (32 waves -> 32 lanes) ---
        if (wv == 0) {
            bv = s_redv[t]; bi = s_redi[t];
#pragma unroll
            for (int off = 16; off > 0; off >>= 1) {
                const float ov = __shfl_xor(bv, off, 32);
                const int   oi = __shfl_xor(bi, off, 32);
                take_better(bv, bi, ov, oi);
            }
            if (t == 0) {
                // Broadcast next farthest point's coords straight into LDS:
                // three async global->LDS b32 copies, tracked by ASYNCcnt.
                const float* ax = px + bi;
                const float* ay = py + bi;
                const float* az = pz + bi;
                unsigned l0 = (unsigned)(size_t)&s_cur[0];
                unsigned l1 = l0 + 4;
                unsigned l2 = l0 + 8;
                asm volatile(
                    "global_load_async_to_lds_b32 %0, %3, off\n\t"
                    "global_load_async_to_lds_b32 %1, %4, off\n\t"
                    "global_load_async_to_lds_b32 %2, %5, off"
                    :
                    : "v"(l0), "v"(l1), "v"(l2), "v"(ax), "v"(ay), "v"(az)
                    : "memory");
            }
        }
        // Async LDS writes must land before the next barrier releases readers.
        asm volatile("s_wait_asynccnt 0x0" ::: "memory");
    }
}

extern "C" void kernel_launch(void* const* d_in, const int* in_sizes, int n_in,
                              void* d_out, int out_size, void* d_ws, size_t ws_size,
                              hipStream_t stream) {
    const float* points = (const float*)d_in[0];   // [B, 3, N] fp32
    float* out = (float*)d_out;                    // [B, 3, S] fp32
    const int B = 8, C = 3;
    const int N = in_sizes[0] / (B * C);           // 131072
    const int S = out_size / (B * C);              // 2048
    fps_kernel<<<dim3(B), dim3(TPB), 0, stream>>>(points, out, N, S);
}